// TopKDecoder_65884798321097
// MI455X (gfx1250) — compile-verified
//
#include <hip/hip_runtime.h>
#include <math.h>

// Problem constants (from reference)
#define B_   32
#define S_   256
#define H_   512
#define V_   10000
#define K_   8
#define T_   32
#define BK_  256          // B_*K_
#define SOS_ 1
#define EOS_ 2
#define NEG_ (-1.0e30f)

#define KC_   16          // K-chunk staged in LDS per pipeline stage
#define LDW_  264         // padded LDS row stride (floats): 16-byte aligned,
                          // row+2 shifts 16 banks -> conflict-free wave reads

typedef __attribute__((ext_vector_type(2))) float v2f;
typedef __attribute__((ext_vector_type(8))) float v8f;
typedef __attribute__((address_space(3))) void lds_void;

// ---------------------------------------------------------------------------
// init: h0 = 0, scores = 0 for beam 0 else NEG, tok = SOS
// ---------------------------------------------------------------------------
__global__ void init_kernel(float* __restrict__ h, float* __restrict__ scores,
                            int* __restrict__ tok) {
  const int i = blockIdx.x * blockDim.x + threadIdx.x;
  if (i < BK_ * H_) h[i] = 0.0f;
  if (i < BK_) {
    scores[i] = ((i % K_) == 0) ? 0.0f : NEG_;
    tok[i] = SOS_;
  }
}

// ---------------------------------------------------------------------------
// attention + context + embedding -> x = [emb | ctx], one block per beam row
// ---------------------------------------------------------------------------
__global__ void attn_embed_kernel(const float* __restrict__ h,
                                  const float* __restrict__ enc,
                                  const float* __restrict__ E,
                                  const int* __restrict__ tok,
                                  float* __restrict__ x) {
  const int r = blockIdx.x;           // beam row 0..255
  const int t = threadIdx.x;          // 256 threads, one per s
  const float* e  = enc + (size_t)(r / K_) * S_ * H_;
  const float* hr = h + (size_t)r * H_;
  __shared__ float att[S_];
  __shared__ float red[256];

  float dot = 0.0f;
  {
    const float* es = e + (size_t)t * H_;
#pragma unroll 8
    for (int j = 0; j < H_; ++j) dot = fmaf(hr[j], es[j], dot);
  }
  att[t] = dot;
  red[t] = dot;
  __syncthreads();
  for (int off = 128; off > 0; off >>= 1) {
    if (t < off) red[t] = fmaxf(red[t], red[t + off]);
    __syncthreads();
  }
  const float m = red[0];
  __syncthreads();
  const float ex = __expf(dot - m);
  red[t] = ex;
  __syncthreads();
  for (int off = 128; off > 0; off >>= 1) {
    if (t < off) red[t] += red[t + off];
    __syncthreads();
  }
  const float inv = 1.0f / red[0];
  att[t] = ex * inv;
  __syncthreads();

  float* xr = x + (size_t)r * (2 * H_);
  for (int j = t; j < H_; j += 256) {
    float c = 0.0f;
    for (int s = 0; s < S_; ++s) c = fmaf(att[s], e[(size_t)s * H_ + j], c);
    xr[H_ + j] = c;                   // ctx in second half
  }
  const float* Er = E + (size_t)tok[r] * H_;
  for (int j = t; j < H_; j += 256) xr[j] = Er[j];   // embedding in first half
}

// ---------------------------------------------------------------------------
// Async staging of one W tile (KC_ rows x 256 cols) into LDS.
// 1024 b128 transfers per tile -> 4 per thread, via the gfx1250
// GLOBAL_LOAD_ASYNC_TO_LDS path (tracked by ASYNCcnt).
// ---------------------------------------------------------------------------
__device__ __forceinline__ void stage_wtile_async(float* dst, const float* __restrict__ W,
                                                  int k0, int nb0, int N, int tid) {
#pragma unroll
  for (int it = 0; it < 4; ++it) {
    const int idx = tid + it * 256;          // float4 index 0..1023
    const int row = idx >> 6;                // 0..15  (64 float4 per row)
    int col = nb0 + ((idx & 63) << 2);
    col = (col <= N - 4) ? col : (N - 4);    // clamp ragged edge (stores guarded)
    const float* gp = W + (size_t)(k0 + row) * N + col;
    const unsigned lds_off =
        (unsigned)(size_t)(lds_void*)(dst + row * LDW_ + ((idx & 63) << 2));
    const unsigned long long ga = (unsigned long long)(size_t)gp;
    asm volatile("global_load_async_to_lds_b128 %0, %1, off"
                 :: "v"(lds_off), "v"(ga) : "memory");
  }
}

// ---------------------------------------------------------------------------
// fp32 WMMA GEMM: C[M,N] = A[M,Kd]*W[Kd,N] (+bias[N]), row-major, M=gridDim.x*32
// 8 waves/block laid out 2(M) x 4(N); wave tile 16x64 (4 accumulators).
// W tiles double-buffered in LDS via async-to-LDS; 16 WMMAs per chunk.
// A fragment (16x4 f32): lanes 0-15 hold {k,k+1}, lanes 16-31 hold {k+2,k+3}.
// C/D (16x16 f32): c[i] -> row m0 + 8*(lane>=16) + i, col n + (lane&15).
// ---------------------------------------------------------------------------
__global__ void gemm_f32_wmma(const float* __restrict__ A,
                              const float* __restrict__ W,
                              const float* __restrict__ bias,
                              float* __restrict__ C,
                              int Kd, int N) {
  __shared__ float wtile[2][KC_ * LDW_];     // ~33 KB double buffer
  const int tid  = threadIdx.x;
  const int lane = tid & 31;
  const int wave = tid >> 5;                 // 0..7
  const int wm   = wave >> 2;                // 0..1
  const int wn   = wave & 3;                 // 0..3
  const int m0   = blockIdx.x * 32 + wm * 16;
  const int nb0  = blockIdx.y * 256;
  const int l16  = lane & 15;
  const int hi   = lane >> 4;                // 0: lanes 0-15, 1: lanes 16-31
  const int jc   = wn * 64 + l16;            // column within the 256-wide tile

  v8f acc[4] = {};
  const float* Ap = A + (size_t)(m0 + l16) * Kd;
  const int nchunks = Kd / KC_;

  stage_wtile_async(wtile[0], W, 0, nb0, N, tid);

  for (int kc = 0; kc < nchunks; ++kc) {
    const float* buf = wtile[kc & 1];
    if (kc + 1 < nchunks) {
      stage_wtile_async(wtile[(kc + 1) & 1], W, (kc + 1) * KC_, nb0, N, tid);
      asm volatile("s_wait_asynccnt 0x4" ::: "memory");   // current tile landed
    } else {
      asm volatile("s_wait_asynccnt 0x0" ::: "memory");
    }
    __syncthreads();                                      // tile visible to all waves

#pragma unroll
    for (int kk = 0; kk < KC_ / 4; ++kk) {
      v2f a;
      const float* ap = Ap + kc * KC_ + kk * 4 + hi * 2;
      a.x = ap[0];
      a.y = ap[1];
      const int r0 = (kk * 4 + hi * 2) * LDW_ + jc;
#pragma unroll
      for (int j = 0; j < 4; ++j) {
        v2f bfrag;
        bfrag.x = buf[r0 + j * 16];
        bfrag.y = buf[r0 + LDW_ + j * 16];
        acc[j] = __builtin_amdgcn_wmma_f32_16x16x4_f32(
            /*neg_a=*/false, a, /*neg_b=*/false, bfrag,
            /*c_mod=*/(short)0, acc[j], /*reuse_a=*/false, /*reuse_b=*/false);
      }
    }
    __syncthreads();                                      // done reading before overwrite
  }

#pragma unroll
  for (int j = 0; j < 4; ++j) {
    const int col = nb0 + jc + j * 16;
    if (col < N) {
      const float bv = bias ? bias[col] : 0.0f;
#pragma unroll
      for (int i = 0; i < 8; ++i) {
        const int row = m0 + hi * 8 + i;
        C[(size_t)row * N + col] = acc[j][i] + bv;
      }
    }
  }
}

// ---------------------------------------------------------------------------
// GRU cell elementwise; writes h_new into h_store slab for step t
// ---------------------------------------------------------------------------
__global__ void gru_kernel(const float* __restrict__ gx, const float* __restrict__ gh,
                           const float* __restrict__ h, float* __restrict__ hnew) {
  const int r = blockIdx.x;
  const float* gxr = gx + (size_t)r * 3 * H_;
  const float* ghr = gh + (size_t)r * 3 * H_;
  for (int j = threadIdx.x; j < H_; j += blockDim.x) {
    const float z  = 1.0f / (1.0f + __expf(-(gxr[j] + ghr[j])));
    const float rr = 1.0f / (1.0f + __expf(-(gxr[H_ + j] + ghr[H_ + j])));
    const float n  = tanhf(gxr[2 * H_ + j] + rr * ghr[2 * H_ + j]);
    hnew[(size_t)r * H_ + j] = (1.0f - z) * n + z * h[(size_t)r * H_ + j];
  }
}

// ---------------------------------------------------------------------------
// topk stage 1: per beam row, log-softmax + (scores + logp) local top-8
// ---------------------------------------------------------------------------
__global__ void topk1_kernel(const float* __restrict__ logits,
                             const float* __restrict__ scores,
                             float* __restrict__ cvals, int* __restrict__ cidx) {
  const int r = blockIdx.x;
  const int t = threadIdx.x;
  const float* L = logits + (size_t)r * V_;
  __shared__ float red[256];
  __shared__ int   redi[256];
  __shared__ float sv[256 * K_];
  __shared__ int   si[256 * K_];

  float m = NEG_;
  for (int v = t; v < V_; v += 256) m = fmaxf(m, L[v]);
  red[t] = m; __syncthreads();
  for (int off = 128; off > 0; off >>= 1) {
    if (t < off) red[t] = fmaxf(red[t], red[t + off]);
    __syncthreads();
  }
  m = red[0]; __syncthreads();
  float s = 0.0f;
  for (int v = t; v < V_; v += 256) s += __expf(L[v] - m);
  red[t] = s; __syncthreads();
  for (int off = 128; off > 0; off >>= 1) {
    if (t < off) red[t] += red[t + off];
    __syncthreads();
  }
  const float lse = m + __logf(red[0]);
  __syncthreads();
  const float base = scores[r] - lse;

  float bv[K_]; int bi[K_];
#pragma unroll
  for (int i = 0; i < K_; ++i) { bv[i] = -3.0e30f; bi[i] = 0; }
  for (int v = t; v < V_; v += 256) {
    const float val = base + L[v];
    if (val > bv[K_ - 1]) {
      int p = K_ - 1;
#pragma unroll
      for (int i = K_ - 2; i >= 0; --i) {
        if (val > bv[i]) { bv[i + 1] = bv[i]; bi[i + 1] = bi[i]; p = i; }
      }
      bv[p] = val; bi[p] = v;
    }
  }
#pragma unroll
  for (int i = 0; i < K_; ++i) { sv[t * K_ + i] = bv[i]; si[t * K_ + i] = bi[i]; }
  __syncthreads();

  for (int sel = 0; sel < K_; ++sel) {
    float lm = -3.0e30f; int li = 0;
#pragma unroll
    for (int i = 0; i < K_; ++i) {
      const float vv = sv[t * K_ + i];
      if (vv > lm) { lm = vv; li = t * K_ + i; }
    }
    red[t] = lm; redi[t] = li; __syncthreads();
    for (int off = 128; off > 0; off >>= 1) {
      if (t < off && red[t + off] > red[t]) { red[t] = red[t + off]; redi[t] = redi[t + off]; }
      __syncthreads();
    }
    if (t == 0) {
      const int w = redi[0];
      cvals[r * K_ + sel] = red[0];
      cidx[r * K_ + sel]  = si[w];
      sv[w] = -3.0e30f;
    }
    __syncthreads();
  }
}

// ---------------------------------------------------------------------------
// topk stage 2 + select: per batch merge 64 candidates -> new beams
// ---------------------------------------------------------------------------
__global__ void topk2_select_kernel(const float* __restrict__ cvals,
                                    const int* __restrict__ cidx,
                                    int* __restrict__ tok, float* __restrict__ scores,
                                    int* __restrict__ pred_t, float* __restrict__ raw_t) {
  const int b = blockIdx.x;
  if (threadIdx.x != 0) return;
  const float* cv = cvals + b * K_ * K_;
  const int*   ci = cidx + b * K_ * K_;
  bool used[K_ * K_];
  for (int i = 0; i < K_ * K_; ++i) used[i] = false;
  for (int sel = 0; sel < K_; ++sel) {
    float best = -4.0e30f; int bj = 0;
    for (int j = 0; j < K_ * K_; ++j)
      if (!used[j] && cv[j] > best) { best = cv[j]; bj = j; }
    used[bj] = true;
    const int kbeam = bj / K_;
    const int v     = ci[bj];
    const int row   = b * K_ + sel;
    tok[row]    = v;
    pred_t[row] = b * K_ + kbeam;
    raw_t[row]  = best;
    scores[row] = (v == EOS_) ? NEG_ : best;
  }
}

// ---------------------------------------------------------------------------
// h <- h_new[pred]
// ---------------------------------------------------------------------------
__global__ void gather_h_kernel(const float* __restrict__ hnew,
                                const int* __restrict__ pred_t,
                                float* __restrict__ h) {
  const int r = blockIdx.x;
  const int src = pred_t[r];
  for (int j = threadIdx.x; j < H_; j += blockDim.x)
    h[(size_t)r * H_ + j] = hnew[(size_t)src * H_ + j];
}

// ---------------------------------------------------------------------------
// backtrack predecessor chain for the top final beam of each batch
// ---------------------------------------------------------------------------
__global__ void backtrack_kernel(const float* __restrict__ raws,
                                 const int* __restrict__ preds,
                                 int* __restrict__ rowsel) {
  const int b = threadIdx.x;
  if (b >= B_) return;
  const float* rl = raws + (size_t)(T_ - 1) * BK_ + b * K_;
  int arg = 0; float best = rl[0];
  for (int k = 1; k < K_; ++k)
    if (rl[k] > best) { best = rl[k]; arg = k; }
  int tp = b * K_ + arg;
  for (int t = T_ - 1; t >= 0; --t) {
    rowsel[t * B_ + b] = tp;
    tp = preds[(size_t)t * BK_ + tp];
  }
}

// ---------------------------------------------------------------------------
// recompute logp for the one selected row per (t, b) -> d_out[T,B,V]
// ---------------------------------------------------------------------------
__global__ void output_kernel(const float* __restrict__ h_store,
                              const int* __restrict__ rowsel,
                              const float* __restrict__ Wo,
                              const float* __restrict__ bo,
                              float* __restrict__ out) {
  const int b = blockIdx.x;
  const int t = blockIdx.y;
  const int tid = threadIdx.x;
  const int row = rowsel[t * B_ + b];
  const float* hr = h_store + (size_t)t * BK_ * H_ + (size_t)row * H_;
  __shared__ float hs[H_];
  __shared__ float lg[V_];      // 40 KB (WGP has 320 KB LDS)
  __shared__ float red[256];
  for (int j = tid; j < H_; j += 256) hs[j] = hr[j];
  __syncthreads();
  for (int v = tid; v < V_; v += 256) {
    float d = bo[v];
#pragma unroll 8
    for (int j = 0; j < H_; ++j) d = fmaf(hs[j], Wo[(size_t)j * V_ + v], d);
    lg[v] = d;
  }
  __syncthreads();
  float m = NEG_;
  for (int v = tid; v < V_; v += 256) m = fmaxf(m, lg[v]);
  red[tid] = m; __syncthreads();
  for (int off = 128; off > 0; off >>= 1) {
    if (tid < off) red[tid] = fmaxf(red[tid], red[tid + off]);
    __syncthreads();
  }
  m = red[0]; __syncthreads();
  float s = 0.0f;
  for (int v = tid; v < V_; v += 256) s += __expf(lg[v] - m);
  red[tid] = s; __syncthreads();
  for (int off = 128; off > 0; off >>= 1) {
    if (tid < off) red[tid] += red[tid + off];
    __syncthreads();
  }
  const float lse = m + __logf(red[0]);
  float* o = out + ((size_t)t * B_ + b) * V_;
  for (int v = tid; v < V_; v += 256) o[v] = lg[v] - lse;
}

// ---------------------------------------------------------------------------
// host driver
// ---------------------------------------------------------------------------
extern "C" void kernel_launch(void* const* d_in, const int* in_sizes, int n_in,
                              void* d_out, int out_size, void* d_ws, size_t ws_size,
                              hipStream_t stream) {
  (void)in_sizes; (void)n_in; (void)out_size; (void)ws_size;
  const float* enc  = (const float*)d_in[0];   // [B,S,H]
  const float* E    = (const float*)d_in[1];   // [V,H]
  const float* Wx   = (const float*)d_in[2];   // [2H,3H]
  const float* Wh   = (const float*)d_in[3];   // [H,3H]
  const float* bias = (const float*)d_in[4];   // [3H]
  const float* Wo   = (const float*)d_in[5];   // [H,V]
  const float* bo   = (const float*)d_in[6];   // [V]
  float* out = (float*)d_out;                  // [T,B,V]

  char* ws = (char*)d_ws;
  size_t off = 0;
  auto alloc = [&](size_t bytes) -> char* {
    char* p = ws + off;
    off = (off + bytes + 255) & ~(size_t)255;
    return p;
  };
  float* h      = (float*)alloc(sizeof(float) * BK_ * H_);
  float* x      = (float*)alloc(sizeof(float) * BK_ * 2 * H_);
  float* gx     = (float*)alloc(sizeof(float) * BK_ * 3 * H_);
  float* gh     = (float*)alloc(sizeof(float) * BK_ * 3 * H_);
  float* logits = (float*)alloc(sizeof(float) * (size_t)BK_ * V_);
  float* scores = (float*)alloc(sizeof(float) * BK_);
  float* cvals  = (float*)alloc(sizeof(float) * BK_ * K_);
  float* hstore = (float*)alloc(sizeof(float) * (size_t)T_ * BK_ * H_);
  float* raws   = (float*)alloc(sizeof(float) * T_ * BK_);
  int*   tok    = (int*)alloc(sizeof(int) * BK_);
  int*   cidx   = (int*)alloc(sizeof(int) * BK_ * K_);
  int*   preds  = (int*)alloc(sizeof(int) * T_ * BK_);
  int*   rowsel = (int*)alloc(sizeof(int) * T_ * B_);

  init_kernel<<<(BK_ * H_ + 255) / 256, 256, 0, stream>>>(h, scores, tok);

  for (int t = 0; t < T_; ++t) {
    float* hnew  = hstore + (size_t)t * BK_ * H_;
    int*   predt = preds + (size_t)t * BK_;
    float* rawt  = raws + (size_t)t * BK_;

    attn_embed_kernel<<<BK_, 256, 0, stream>>>(h, enc, E, tok, x);
    gemm_f32_wmma<<<dim3(BK_ / 32, (3 * H_ + 255) / 256), 256, 0, stream>>>(
        x, Wx, bias, gx, 2 * H_, 3 * H_);
    gemm_f32_wmma<<<dim3(BK_ / 32, (3 * H_ + 255) / 256), 256, 0, stream>>>(
        h, Wh, nullptr, gh, H_, 3 * H_);
    gru_kernel<<<BK_, 256, 0, stream>>>(gx, gh, h, hnew);
    gemm_f32_wmma<<<dim3(BK_ / 32, (V_ + 255) / 256), 256, 0, stream>>>(
        hnew, Wo, bo, logits, H_, V_);
    topk1_kernel<<<BK_, 256, 0, stream>>>(logits, scores, cvals, cidx);
    topk2_select_kernel<<<B_, 32, 0, stream>>>(cvals, cidx, tok, scores, predt, rawt);
    gather_h_kernel<<<BK_, 256, 0, stream>>>(hnew, predt, h);
  }

  backtrack_kernel<<<1, 32, 0, stream>>>(raws, preds, rowsel);
  output_kernel<<<dim3(B_, T_), 256, 0, stream>>>(hstore, rowsel, Wo, bo, out);
}